// IAF_88244398063690
// MI455X (gfx1250) — compile-verified
//
#include <hip/hip_runtime.h>
#include <math.h>

namespace {
constexpr int kB = 16384;
constexpr int kD = 64;
constexpr int kWaves = 4;
constexpr int kThreads = kWaves * 32;          // wave32: 4 waves/block
constexpr int kRowsPerWave = 16;               // one 16-row WMMA strip per wave
constexpr int kRowsPerBlock = kWaves * kRowsPerWave;
constexpr int kAst = 68;                       // activation LDS row stride (floats): bank-conflict pad
constexpr int kWPS = 160;                      // weight K-pair row stride (floats): (160 mod 64)=32
                                               //  -> lane-halves split banks 0-31 / 32-63, conflict-free
constexpr float kLog2Pi = 1.8378770664093453f;
}

typedef float v2f __attribute__((ext_vector_type(2)));
typedef float v8f __attribute__((ext_vector_type(8)));

// Wave-private LDS ordering: DS ops of one wave are in-order in HW; we only
// need to stop the compiler from reordering across the exchange point.
__device__ __forceinline__ void wave_fence() { asm volatile("" ::: "memory"); }

__global__ void __launch_bounds__(kThreads)
iaf_kernel(const float* __restrict__ zmean, const float* __restrict__ zlogvar,
           const float* __restrict__ noise, const float* __restrict__ Wg,
           const float* __restrict__ bvec, float* __restrict__ out_samples,
           float* __restrict__ out_lp)
{
  // Masked weights, K-pair interleaved: element (l,k,n) at
  //   (l*32 + k/2)*kWPS + 2*n + (k&1)
  // so a lane's WMMA B-fragment {W[k][c], W[k+1][c]} is one 8-byte DS read.
  __shared__ float s_w[4 * 32 * kWPS];                    // 81920 B
  __shared__ float s_b[4 * kD];
  __shared__ float s_act[kWaves][2][kRowsPerWave * kAst]; // wave-private ping-pong tiles
  __shared__ float s_red[kThreads * 8];                   // lp reduction scratch

  const int tid = threadIdx.x;
  const int lane = tid & 31;
  const int wv = tid >> 5;
  const int half = lane >> 4;     // 0: lanes 0-15 (rows 0-7 / K 0-1), 1: lanes 16-31 (rows 8-15 / K 2-3)
  const int l15 = lane & 15;
  const int r0 = blockIdx.x * kRowsPerBlock + wv * kRowsPerWave;

  // ---- cooperative masked-weight load for stage s ----
  auto load_weights = [&](int s) {
    __syncthreads();
    for (int idx = tid; idx < 4 * kD * kD; idx += kThreads) {
      const int l = idx >> 12;          // layer
      const int rem = idx & 4095;
      const int k = rem >> 6;           // input index (K)
      const int n = rem & 63;           // output index (N)
      const bool keep = (l == 0) ? (k < n) : (k <= n);   // excl / incl triangular masks
      s_w[(l * 32 + (k >> 1)) * kWPS + 2 * n + (k & 1)] =
          keep ? Wg[((s * 4 + l) * kD + k) * kD + n] : 0.0f;
    }
    for (int idx = tid; idx < 4 * kD; idx += kThreads)
      s_b[idx] = bvec[s * 4 * kD + idx];
    __syncthreads();
  };

  // ---- one masked dense layer: [relu](in @ Wl + bl) [+ addfr] over the wave's 16x64 tile ----
  // ib/ob: ping-pong buffer indices into s_act[wv][]; ob < 0 => return fragments in outfr.
  // addfr != nullptr: add fragments before the LDS store (density fixed-point fusion).
  auto layer = [&](int ib, int ob, int l, bool do_relu, v8f* outfr, const v8f* addfr) {
    v8f acc[4];
#pragma unroll
    for (int n = 0; n < 4; ++n) {
      const float bv = s_b[l * kD + n * 16 + l15];
      v8f t;
#pragma unroll
      for (int v = 0; v < 8; ++v) t[v] = bv;
      acc[n] = t;
    }
    const int arow = l15 * kAst;
    const int wbase = 2 * l15;          // per-lane part of the weight-pair address
#pragma unroll
    for (int kk = 0; kk < 16; ++kk) {   // fully unrolled: LDS offsets become immediates
      const int kb = kk * 4 + half * 2;
      // A fragment: 16x4 fp32, lane row = l15, K pair by lane half (adjacent floats)
      v2f a;
      a.x = s_act[wv][ib][arow + kb];
      a.y = s_act[wv][ib][arow + kb + 1];
      const int pbase = (l * 32 + kk * 2 + half) * kWPS + wbase;
#pragma unroll
      for (int n = 0; n < 4; ++n) {
        // B fragment: 4x16 fp32, lane col = l15, K pair pre-interleaved (adjacent floats)
        v2f bf;
        bf.x = s_w[pbase + 32 * n];
        bf.y = s_w[pbase + 32 * n + 1];
        acc[n] = __builtin_amdgcn_wmma_f32_16x16x4_f32(
            false, a, false, bf, (short)0, acc[n], false, false);
      }
    }
    if (do_relu) {
#pragma unroll
      for (int n = 0; n < 4; ++n)
#pragma unroll
        for (int v = 0; v < 8; ++v) acc[n][v] = fmaxf(acc[n][v], 0.0f);
    }
    if (addfr) {
#pragma unroll
      for (int n = 0; n < 4; ++n)
#pragma unroll
        for (int v = 0; v < 8; ++v) acc[n][v] += addfr[n][v];
    }
    if (ob >= 0) {
#pragma unroll
      for (int n = 0; n < 4; ++n)
#pragma unroll
        for (int v = 0; v < 8; ++v)
          s_act[wv][ob][(v + half * 8) * kAst + n * 16 + l15] = acc[n][v];
      wave_fence();
    } else {
#pragma unroll
      for (int n = 0; n < 4; ++n) outfr[n] = acc[n];
    }
  };

  // write register C-fragments into LDS tile `b` (C/D layout -> row major)
  auto store_frags = [&](const v8f* fr, int b) {
#pragma unroll
    for (int n = 0; n < 4; ++n)
#pragma unroll
      for (int v = 0; v < 8; ++v)
        s_act[wv][b][(v + half * 8) * kAst + n * 16 + l15] = fr[n][v];
    wave_fence();
  };

  // made(): input tile must already be in buffer 0; result in outfr (no relu on last layer)
  auto made = [&](v8f* outfr) {
    layer(0, 1, 0, true, nullptr, nullptr);
    layer(1, 0, 1, true, nullptr, nullptr);
    layer(0, 1, 2, true, nullptr, nullptr);
    layer(1, -1, 3, false, outfr, nullptr);
  };

  auto reverse_cols = [&](v8f* fr) {   // fr[.,col] <- fr[.,63-col] via LDS
    store_frags(fr, 0);
#pragma unroll
    for (int n = 0; n < 4; ++n)
#pragma unroll
      for (int v = 0; v < 8; ++v)
        fr[n][v] = s_act[wv][0][(v + half * 8) * kAst + (63 - (n * 16 + l15))];
    wave_fence();
  };

  // ---------------- eps = mean + exp(0.5*logvar)*noise ----------------
  v8f y[4];
#pragma unroll
  for (int n = 0; n < 4; ++n)
#pragma unroll
    for (int v = 0; v < 8; ++v) {
      const int row = r0 + v + half * 8;
      const int col = n * 16 + l15;
      const float m = zmean[row * kD + col];
      const float lv = zlogvar[row * kD + col];
      const float e = noise[row * kD + col];
      y[n][v] = m + expf(0.5f * lv) * e;
    }

  v8f m4[4];

  // ---------------- sampling: s = 1 then s = 0 ----------------
  load_weights(1);
  store_frags(y, 0);
  made(m4);
#pragma unroll
  for (int n = 0; n < 4; ++n)
#pragma unroll
    for (int v = 0; v < 8; ++v) y[n][v] -= m4[n][v];
  reverse_cols(y);

  load_weights(0);
  store_frags(y, 0);
  made(m4);
#pragma unroll
  for (int n = 0; n < 4; ++n)
#pragma unroll
    for (int v = 0; v < 8; ++v) y[n][v] -= m4[n][v];

  // samples output
#pragma unroll
  for (int n = 0; n < 4; ++n)
#pragma unroll
    for (int v = 0; v < 8; ++v)
      out_samples[(r0 + v + half * 8) * kD + n * 16 + l15] = y[n][v];

  // ---------------- density: 64-iteration fixed point per stage ----------------
  // u lives in LDS buffer 0; the last layer of each iteration writes x0 + made(u)
  // straight back into buffer 0 (no separate store pass, no register round-trip).
  auto density_stage = [&]() {
    v8f x0[4];
#pragma unroll
    for (int n = 0; n < 4; ++n) x0[n] = y[n];
    // u <- 0
#pragma unroll
    for (int n = 0; n < 4; ++n)
#pragma unroll
      for (int v = 0; v < 8; ++v)
        s_act[wv][0][(v + half * 8) * kAst + n * 16 + l15] = 0.0f;
    wave_fence();
    for (int it = 0; it < kD; ++it) {   // fori_loop(0, D, u <- x0 + made(u))
      layer(0, 1, 0, true, nullptr, nullptr);
      layer(1, 0, 1, true, nullptr, nullptr);
      layer(0, 1, 2, true, nullptr, nullptr);
      layer(1, 0, 3, false, nullptr, x0);   // fused: buf0 <- x0 + out
    }
#pragma unroll
    for (int n = 0; n < 4; ++n)
#pragma unroll
      for (int v = 0; v < 8; ++v)
        y[n][v] = s_act[wv][0][(v + half * 8) * kAst + n * 16 + l15];
    wave_fence();
  };

  density_stage();          // stage 0 (weights already loaded)
  reverse_cols(y);
  load_weights(1);
  density_stage();          // stage 1

  // ---------------- lp = -0.5 * sum_col((x-m)^2 e^{-lv} + lv + log2pi) ----------------
  float part[8];
#pragma unroll
  for (int v = 0; v < 8; ++v) part[v] = 0.0f;
#pragma unroll
  for (int n = 0; n < 4; ++n)
#pragma unroll
    for (int v = 0; v < 8; ++v) {
      const int row = r0 + v + half * 8;
      const int col = n * 16 + l15;
      const float m = zmean[row * kD + col];
      const float lv = zlogvar[row * kD + col];
      const float d = y[n][v] - m;
      part[v] += d * d * expf(-lv) + lv + kLog2Pi;
    }
#pragma unroll
  for (int v = 0; v < 8; ++v) s_red[tid * 8 + v] = part[v];
  wave_fence();
  if (lane < 16) {                        // one lane per row of this wave's strip
    const int h2 = lane >> 3;
    const int vv = lane & 7;
    const int base = wv * 32 + h2 * 16;
    float s = 0.0f;
#pragma unroll
    for (int j = 0; j < 16; ++j) s += s_red[(base + j) * 8 + vv];
    out_lp[r0 + lane] = -0.5f * s;
  }
}

extern "C" void kernel_launch(void* const* d_in, const int* in_sizes, int n_in,
                              void* d_out, int out_size, void* d_ws, size_t ws_size,
                              hipStream_t stream) {
  (void)in_sizes; (void)n_in; (void)out_size; (void)d_ws; (void)ws_size;
  const float* zmean = (const float*)d_in[0];
  const float* zlogvar = (const float*)d_in[1];
  const float* noise = (const float*)d_in[2];
  const float* Wg = (const float*)d_in[3];
  const float* bvec = (const float*)d_in[4];
  float* samples = (float*)d_out;
  float* lp = samples + (size_t)kB * kD;
  const dim3 grid(kB / kRowsPerBlock);   // 256 blocks x 128 threads (4 waves)
  iaf_kernel<<<grid, kThreads, 0, stream>>>(zmean, zlogvar, noise, Wg, bvec, samples, lp);
}